// HEALPixAttention_9294309228546
// MI455X (gfx1250) — compile-verified
//
#include <hip/hip_runtime.h>
#include <hip/hip_bf16.h>
#include <math.h>

typedef __attribute__((ext_vector_type(2))) float v2f;
typedef __attribute__((ext_vector_type(8))) float v8f;

#define N_SEQ 2048
#define EMB   512
#define NH    8
#define HD    64
#define QKVW  1536   // 3*EMB
#define KK    512    // reduction dim of both dense GEMMs

// ---- CDNA5 async global->LDS staging (ASYNCcnt path), with safe fallback ----
#if defined(__HIP_DEVICE_COMPILE__) && __has_builtin(__builtin_amdgcn_global_load_async_to_lds_b128)
#define HAVE_ASYNC_LDS 1
#else
#define HAVE_ASYNC_LDS 0
#endif

#if HAVE_ASYNC_LDS
typedef int v4i_gcc __attribute__((vector_size(16)));
__device__ __forceinline__ void async_copy16(const float* g, float* l) {
    __builtin_amdgcn_global_load_async_to_lds_b128(
        (__attribute__((address_space(1))) v4i_gcc*)g,
        (__attribute__((address_space(3))) v4i_gcc*)l,
        0, 0);
}
__device__ __forceinline__ void async_wait0() {
#if __has_builtin(__builtin_amdgcn_s_wait_asynccnt)
    __builtin_amdgcn_s_wait_asynccnt(0);
#else
    asm volatile("s_wait_asynccnt 0" ::: "memory");
#endif
}
#endif

// ---------------------------------------------------------------------------
// Row-major f32 WMMA GEMM with K=512:  C[M,N] = A[M,512] @ B[512,N] + bias[N]
// Block = 256 threads = 8 waves; each wave -> one 16x16 tile.
// A-tile (16x512, shared by all 8 waves) staged to LDS via async-to-LDS.
// ---------------------------------------------------------------------------
__global__ __launch_bounds__(256) void wmma_gemm_k512(
    const float* __restrict__ A, const float* __restrict__ B,
    const float* __restrict__ bias, float* __restrict__ C,
    int M, int N)
{
    __shared__ float Alds[16 * KK];          // 32 KB

    const int tid  = threadIdx.x;
    const int wave = tid >> 5;
    const int lane = tid & 31;
    const int half = lane >> 4;              // 0: K pair {0,1}, 1: K pair {2,3}
    const int r    = lane & 15;

    const int n0 = (blockIdx.x * 8 + wave) * 16;
    const int m0 = blockIdx.y * 16;

    // stage A tile once per block (8x reuse across waves)
#if HAVE_ASYNC_LDS
    for (int i = tid; i < (16 * KK) / 4; i += 256) {
        const int row = i >> 7;              // (i*4)/512
        const int col = (i & 127) * 4;       // (i*4)%512
        async_copy16(A + (size_t)(m0 + row) * KK + col, Alds + row * KK + col);
    }
    async_wait0();
#else
    for (int i = tid; i < 16 * KK; i += 256)
        Alds[i] = A[(size_t)(m0 + (i >> 9)) * KK + (i & (KK - 1))];
#endif
    __syncthreads();

    v8f acc = {};
    const float* Ar = Alds + r * KK;
    for (int k = 0; k < KK; k += 4) {
        v2f a, b;
        a.x = Ar[k + 2 * half];
        a.y = Ar[k + 2 * half + 1];
        const float* Bp = B + (size_t)(k + 2 * half) * N + n0 + r;
        b.x = Bp[0];
        b.y = Bp[N];
        acc = __builtin_amdgcn_wmma_f32_16x16x4_f32(
            false, a, false, b, (short)0, acc, false, false);
    }
    const int col = n0 + r;
    const float bv = bias[col];
#pragma unroll
    for (int i = 0; i < 8; ++i) {
        const int row = m0 + i + 8 * half;   // C VGPR i -> M = i + 8*half
        C[(size_t)row * N + col] = acc[i] + bv;
    }
}

// ---------------------------------------------------------------------------
// Fused attention for one (16-query tile, head):
//   S = (Q @ K^T)/8   (WMMA; 16x2048 strip resident in LDS — 320KB-LDS design)
//   P = softmax_k(S)  (in place)
//   P += geodesic bias: 1 acos + 1 sincos + 32-step harmonic recurrence
//   y[16,64] = P @ V_head (WMMA, A-frags from LDS)
// ---------------------------------------------------------------------------
__global__ __launch_bounds__(256) void attn_fused(
    const float* __restrict__ qkv,      // [N_SEQ, 1536]: per row h*192 + {0:q,64:k,128:v}
    const float* __restrict__ vec,      // [N_SEQ, 3]
    const float* __restrict__ Wbias,    // [64, 8]
    const float* __restrict__ bbias,    // [8]
    float* __restrict__ yattn)          // [N_SEQ, 512]
{
    extern __shared__ float smem[];
    float* S    = smem;                  // 16 * 2048  (128 KB)
    float* Qs   = smem + 16 * 2048;      // 16 * 64
    float* Yp   = Qs   + 16 * 64;        // 16 * 64
    float* WB   = Yp   + 16 * 64;        // 64
    float* RED  = WB   + 64;             // 256
    float* ROWM = RED  + 256;            // 16
    float* ROWZ = ROWM + 16;             // 16

    const int qtile = blockIdx.x;        // 0..127
    const int h     = blockIdx.y;        // 0..7
    const int tid   = threadIdx.x;
    const int wave  = tid >> 5;
    const int lane  = tid & 31;
    const int half  = lane >> 4;
    const int r     = lane & 15;

    // --- stage raw Q tile (16x64) via async-to-LDS; W_bias column h ---
#if HAVE_ASYNC_LDS
    {
        const int row  = tid >> 4;
        const int col4 = (tid & 15) * 4;
        async_copy16(qkv + (size_t)(qtile * 16 + row) * QKVW + h * 192 + col4,
                     Qs + row * 64 + col4);
    }
    async_wait0();
#else
    for (int idx = tid; idx < 16 * 64; idx += 256)
        Qs[idx] = qkv[(size_t)(qtile * 16 + (idx >> 6)) * QKVW + h * 192 + (idx & 63)];
#endif
    for (int f = tid; f < 64; f += 256) WB[f] = Wbias[f * NH + h];
    __syncthreads();

    // --- S = (Q @ K^T)/8 : 128 key-tiles, wave w takes kt = w, w+8, ... ---
    for (int kt = wave; kt < N_SEQ / 16; kt += 8) {
        v8f acc = {};
        const float* Krow = qkv + (size_t)(kt * 16 + r) * QKVW + h * 192 + 64;
#pragma unroll
        for (int kk = 0; kk < HD; kk += 4) {
            v2f a, b;
            a.x = Qs[r * 64 + kk + 2 * half];
            a.y = Qs[r * 64 + kk + 2 * half + 1];
            b.x = Krow[kk + 2 * half];       // B[k, n] = K[n, k]
            b.y = Krow[kk + 2 * half + 1];
            acc = __builtin_amdgcn_wmma_f32_16x16x4_f32(
                false, a, false, b, (short)0, acc, false, false);
        }
#pragma unroll
        for (int i = 0; i < 8; ++i)
            S[(i + 8 * half) * N_SEQ + kt * 16 + r] = acc[i] * 0.125f;  // 1/sqrt(HD)
    }
    __syncthreads();

    // --- softmax over keys : thread -> (row q = tid>>4, slice sub = tid&15) ---
    const int q   = tid >> 4;
    const int sub = tid & 15;
    float* Srow = S + q * N_SEQ;

    float m = -3.402823466e+38f;
    for (int j = 0; j < N_SEQ / 16; ++j) m = fmaxf(m, Srow[sub + 16 * j]);
    RED[tid] = m;
    __syncthreads();
    if (sub == 0) {
        float mm = RED[q * 16];
        for (int i = 1; i < 16; ++i) mm = fmaxf(mm, RED[q * 16 + i]);
        ROWM[q] = mm;
    }
    __syncthreads();
    m = ROWM[q];
    float z = 0.f;
    for (int j = 0; j < N_SEQ / 16; ++j) {
        const int kcol = sub + 16 * j;
        const float e = __expf(Srow[kcol] - m);
        Srow[kcol] = e;
        z += e;
    }
    RED[tid] = z;
    __syncthreads();
    if (sub == 0) {
        float zz = 0.f;
        for (int i = 0; i < 16; ++i) zz += RED[q * 16 + i];
        ROWZ[q] = zz;
    }
    __syncthreads();
    const float invz = 1.0f / ROWZ[q];

    // --- normalize + post-softmax geodesic bias (harmonic recurrence) ---
    {
        const float* vq = vec + (size_t)(qtile * 16 + q) * 3;
        const float vqx = vq[0], vqy = vq[1], vqz = vq[2];
        const float bb = bbias[h];
        for (int j = 0; j < N_SEQ / 16; ++j) {
            const int k = sub + 16 * j;
            const float* vk = vec + (size_t)k * 3;
            float d = vqx * vk[0] + vqy * vk[1] + vqz * vk[2];
            d = fminf(fmaxf(d, 0.0f), 1.0f);
            // angle = acos(d)/0.001 ; freqs = linspace(0,1,32)/0.1 = f/3.1
            const float theta = acosf(d) * (1000.0f * 0.3225806451612903f);
            float s1, c1;
            sincosf(theta, &s1, &c1);
            float sf = 0.0f, cf = 1.0f;      // sin/cos of f*theta, f=0
            float bacc = 0.0f;
#pragma unroll
            for (int f = 0; f < 32; ++f) {
                bacc = fmaf(sf, WB[f], bacc);
                bacc = fmaf(cf, WB[32 + f], bacc);
                const float sn = fmaf(sf, c1, cf * s1);
                const float cn = fmaf(cf, c1, -sf * s1);
                sf = sn; cf = cn;
            }
            Srow[k] = Srow[k] * invz + bacc + bb;
        }
    }
    __syncthreads();

    // --- y[16,64] = P @ V_head : 8 waves = 4 n-tiles x 2 K-halves ---
    const int ntile = wave & 3;
    const int khalf = wave >> 2;
    const int n0 = ntile * 16;
    v8f acc = {};
    const int kbeg = khalf * (N_SEQ / 2), kend = kbeg + N_SEQ / 2;
    for (int kk = kbeg; kk < kend; kk += 4) {
        const int krow = kk + 2 * half;
        v2f a, b;
        a.x = S[r * N_SEQ + krow];
        a.y = S[r * N_SEQ + krow + 1];
        const float* Vp = qkv + (size_t)krow * QKVW + h * 192 + 128 + n0 + r;
        b.x = Vp[0];
        b.y = Vp[QKVW];
        acc = __builtin_amdgcn_wmma_f32_16x16x4_f32(
            false, a, false, b, (short)0, acc, false, false);
    }
    if (khalf == 1) {
#pragma unroll
        for (int i = 0; i < 8; ++i)
            Yp[(i + 8 * half) * 64 + n0 + r] = acc[i];
    }
    __syncthreads();
    if (khalf == 0) {
#pragma unroll
        for (int i = 0; i < 8; ++i) {
            const int row = i + 8 * half;
            const float val = acc[i] + Yp[row * 64 + n0 + r];
            yattn[(size_t)(qtile * 16 + row) * EMB + h * HD + n0 + r] = val;
        }
    }
}

// ---------------------------------------------------------------------------
extern "C" void kernel_launch(void* const* d_in, const int* in_sizes, int n_in,
                              void* d_out, int out_size, void* d_ws, size_t ws_size,
                              hipStream_t stream) {
    (void)in_sizes; (void)n_in; (void)out_size; (void)ws_size;
    const float* x      = (const float*)d_in[0];   // [1,2048,512]
    const float* vec    = (const float*)d_in[1];   // [1,2048,3]
    const float* W_qkv  = (const float*)d_in[2];   // [512,1536]
    const float* b_qkv  = (const float*)d_in[3];   // [1536]
    const float* W_bias = (const float*)d_in[4];   // [64,8]
    const float* b_bias = (const float*)d_in[5];   // [8]
    const float* W_out  = (const float*)d_in[6];   // [512,512]
    const float* b_out  = (const float*)d_in[7];   // [512]

    float* qkv   = (float*)d_ws;                       // 2048*1536 f32 (12.6 MB)
    float* yattn = qkv + (size_t)N_SEQ * QKVW;         // 2048*512  f32 (4.2 MB)

    const dim3 blk(256);

    // K1: qkv = x @ W_qkv + b_qkv    (tiles: 128 M x 96 N)
    wmma_gemm_k512<<<dim3(QKVW / 16 / 8, N_SEQ / 16), blk, 0, stream>>>(
        x, W_qkv, b_qkv, qkv, N_SEQ, QKVW);

    // K2: fused attention + geodesic bias, one block per (16 queries, head)
    const size_t smem_bytes =
        (16 * 2048 + 16 * 64 + 16 * 64 + 64 + 256 + 16 + 16) * sizeof(float); // ~137.5 KB
    (void)hipFuncSetAttribute((const void*)attn_fused,
                              hipFuncAttributeMaxDynamicSharedMemorySize,
                              (int)smem_bytes);
    attn_fused<<<dim3(N_SEQ / 16, NH), blk, smem_bytes, stream>>>(
        qkv, vec, W_bias, b_bias, yattn);

    // K3: out = yattn @ W_out + b_out  (tiles: 128 M x 32 N)
    wmma_gemm_k512<<<dim3(EMB / 16 / 8, N_SEQ / 16), blk, 0, stream>>>(
        yattn, W_out, b_out, (float*)d_out, N_SEQ, EMB);
}